// Loss_mv_ms_49538152792975
// MI455X (gfx1250) — compile-verified
//
#include <hip/hip_runtime.h>
#include <math.h>

// CDNA5 / gfx1250 fused "attention-SVM" loss.
//   For each view X in {fi, fj[1], fj[2]}:
//     U[i] = sum_j exp(fi[i].X[j]) * X[j]        (softmax denom cancels under row-normalize)
//     S_v  = mean_i  (fi[i].U[i]) / max(||U[i]||, 1e-12)
//   loss = 3 * [ (1/a)log1p(exp(-a(S_0-l))) + (1/b)log1p(exp(b(S_1-l)) + exp(b(S_1+S_2-l))) ]
// Scores are cosines in [-1,1] (rows L2-normalized) -> exp is always safe, no online softmax.
// K tiles are double-buffered in LDS and fetched with GLOBAL_LOAD_ASYNC_TO_LDS_B128
// (ASYNCcnt-tracked DMA, no VGPR bounce) so global latency hides under the WMMA work.

typedef __attribute__((ext_vector_type(2))) float v2f;
typedef __attribute__((ext_vector_type(8))) float v8f;

constexpr int B_    = 8192;
constexpr int D_    = 128;
constexpr int QSTR  = 132;   // Q LDS row stride (floats): conflict-free b64 A loads, 16B aligned
constexpr int KSTR  = 136;   // K LDS row stride (floats): conflict-free 2nd-GEMM B loads, 16B aligned
constexpr int PSTR  = 20;    // P LDS row stride (floats)
constexpr int WAVES = 4;     // wave32s per workgroup
constexpr int NVIEW = 3;
constexpr int ROWS_PER_WG = WAVES * 16;

// Issue one 16x128 K tile as 16 wave-wide async-to-LDS b128 transfers (4 per wave).
__device__ __forceinline__ void async_stage_tile(const float* __restrict__ X, int kb,
                                                 float* __restrict__ dstbase, int tid)
{
    const int r  = tid >> 3;          // 0..15 : key row within tile
    const int c0 = (tid & 7) * 16;    // 0..112: starting column
    const float* src = X + (size_t)(kb + r) * D_ + c0;
    float*       dst = dstbase + r * KSTR + c0;
    #pragma unroll
    for (int u = 0; u < 4; ++u) {
        const unsigned long long g = (unsigned long long)(uintptr_t)(src + 4 * u);
        const unsigned int       l = (unsigned int)(uintptr_t)(dst + 4 * u); // flat[31:0] == LDS addr
        asm volatile("global_load_async_to_lds_b128 %0, %1, off"
                     :: "v"(l), "v"(g) : "memory");
    }
}

__global__ __launch_bounds__(WAVES * 32)
void attn_svm_kernel(const float* __restrict__ fi, const float* __restrict__ fj,
                     float* __restrict__ partials, int nwg)
{
    __shared__ __align__(16) float q_lds[WAVES * 16 * QSTR]; // 33.8 KB
    __shared__ __align__(16) float k_lds[2][16 * KSTR];      // 17.4 KB double-buffered K tile
    __shared__ __align__(16) float p_lds[WAVES * 16 * PSTR]; //  5.1 KB

    const int tid  = threadIdx.x;
    const int wave = tid >> 5;
    const int lane = tid & 31;
    const int half = lane >> 4;   // WMMA half-wave (K-split / M-split)
    const int ln   = lane & 15;   // M (A-matrix) or N (B/C/D) index

    const int view = blockIdx.y;  // 0 -> fi, 1 -> fj[1], 2 -> fj[2]
    const float* X = (view == 0) ? fi : (fj + (size_t)view * (size_t)(B_ * D_));

    const int qrow0 = blockIdx.x * ROWS_PER_WG + wave * 16;
    float* qw = q_lds + wave * 16 * QSTR;
    float* pw = p_lds + wave * 16 * PSTR;

    // Stage this wave's 16 query rows (always fi) into LDS; batch 4 loads per group
    // through distinct temporaries so the global loads overlap.
    #pragma unroll
    for (int r4 = 0; r4 < 16; r4 += 4) {
        float4 t[4];
        #pragma unroll
        for (int r = 0; r < 4; ++r)
            t[r] = *(const float4*)(fi + (size_t)(qrow0 + r4 + r) * D_ + lane * 4);
        #pragma unroll
        for (int r = 0; r < 4; ++r)
            *(float4*)(qw + (r4 + r) * QSTR + lane * 4) = t[r];
    }

    // U accumulators: 16x128 f32 tile = 8 column-blocks of 16x16 (8 VGPRs each).
    v8f U[8];
    #pragma unroll
    for (int d = 0; d < 8; ++d) U[d] = {};

    // Prime the async pipeline with tile 0.
    async_stage_tile(X, 0, &k_lds[0][0], tid);
    int cur = 0;

    for (int kb = 0; kb < B_; kb += 16) {
        // My async loads for buffer `cur` have landed in LDS...
        asm volatile("s_wait_asynccnt 0x0" ::: "memory");
        // ...and after the barrier, everyone's have; nobody still reads buffer `cur^1`.
        __syncthreads();
        if (kb + 16 < B_)
            async_stage_tile(X, kb + 16, &k_lds[cur ^ 1][0], tid);   // prefetch next tile
        const float* kt = &k_lds[cur][0];

        // ---- Score GEMM: S(16x16) = Q(16x128) . K^T via 32x V_WMMA_F32_16X16X4_F32.
        // Two accumulator chains for matrix-pipe ILP; merged in the exp below.
        // A lane layout: a = Q[ln][4c + 2*half + {0,1}]   (16x4 A-matrix)
        // B lane layout: b = K^T[2*half + {0,1}][ln] = K[ln][4c + 2*half + {0,1}]
        v8f acc0 = {}, acc1 = {};
        #pragma unroll
        for (int c = 0; c < 32; c += 2) {
            const int k0 = 4 * c + 2 * half;
            v2f a0 = *(const v2f*)(qw + ln * QSTR + k0);
            v2f b0 = *(const v2f*)(kt + ln * KSTR + k0);
            acc0 = __builtin_amdgcn_wmma_f32_16x16x4_f32(
                false, a0, false, b0, (short)0, acc0, false, false);
            v2f a1 = *(const v2f*)(qw + ln * QSTR + k0 + 4);
            v2f b1 = *(const v2f*)(kt + ln * KSTR + k0 + 4);
            acc1 = __builtin_amdgcn_wmma_f32_16x16x4_f32(
                false, a1, false, b1, (short)0, acc1, false, false);
        }

        // exp of scores (|score| <= 1, safe). D-layout: p[v] = S[v + 8*half][ln].
        v8f p;
        #pragma unroll
        for (int v = 0; v < 8; ++v) p[v] = __expf(acc0[v] + acc1[v]);

        // Bounce P through LDS: D-layout (lane=N, vgpr=M) -> row-major -> A-layout loads.
        #pragma unroll
        for (int v = 0; v < 8; ++v)
            pw[(v + 8 * half) * PSTR + ln] = p[v];
        // same-wave LDS ops are in-order; compiler inserts the ds wait.

        // ---- Accumulate GEMM: U(16x128) += P(16x16) . K(16x128), 32x WMMA f32
        // (8-way ILP across the independent U[db] chains).
        #pragma unroll
        for (int cc = 0; cc < 4; ++cc) {
            const int kk = 4 * cc + 2 * half;
            v2f a = *(const v2f*)(pw + ln * PSTR + kk);   // A = P[ln][kk..kk+1], reused x8
            #pragma unroll
            for (int db = 0; db < 8; ++db) {
                v2f b;                                     // B = K[kk..kk+1][db*16 + ln]
                b.x = kt[(kk    ) * KSTR + db * 16 + ln];
                b.y = kt[(kk + 1) * KSTR + db * 16 + ln];
                U[db] = __builtin_amdgcn_wmma_f32_16x16x4_f32(
                    false, a, false, b, (short)0, U[db], false, false);
            }
        }
        cur ^= 1;
    }

    // ---- Epilogue: per row m = v + 8*half: ||U[m]||^2 and fi[m].U[m], reduce over 16 lanes (N).
    float contrib = 0.f;
    #pragma unroll
    for (int v = 0; v < 8; ++v) {
        const int m = v + 8 * half;
        float ssq = 0.f, sdot = 0.f;
        #pragma unroll
        for (int db = 0; db < 8; ++db) {
            const float u = U[db][v];
            const float q = qw[m * QSTR + db * 16 + ln];
            ssq  += u * u;
            sdot += u * q;
        }
        #pragma unroll
        for (int off = 8; off >= 1; off >>= 1) {
            ssq  += __shfl_xor(ssq,  off, 32);
            sdot += __shfl_xor(sdot, off, 32);
        }
        contrib += sdot / fmaxf(sqrtf(ssq), 1e-12f);  // sum of 8 rows' fi.Vn
    }
    if (ln == 0) {   // one deterministic partial per half-wave (covers rows 8*half..8*half+7)
        const int pid = (view * nwg + blockIdx.x) * (WAVES * 2) + wave * 2 + half;
        partials[pid] = contrib;
    }
}

// Reduce 1024 partials per view, then apply the scalar loss tail in double precision.
__global__ __launch_bounds__(1024)
void reduce_finalize_kernel(const float* __restrict__ partials,
                            float* __restrict__ out, int per_view)
{
    __shared__ float  sm[1024];
    __shared__ double Sv[NVIEW];
    const int t = threadIdx.x;
    for (int v = 0; v < NVIEW; ++v) {
        sm[t] = (t < per_view) ? partials[v * per_view + t] : 0.f;
        __syncthreads();
        for (int s = 512; s > 0; s >>= 1) {
            if (t < s) sm[t] += sm[t + s];
            __syncthreads();
        }
        if (t == 0) Sv[v] = (double)sm[0] / (double)B_;
        __syncthreads();
    }
    if (t == 0) {
        const double ALPHA = 1.5, BETA = 45.0, LAMDA = 0.5;
        const double smm1 = Sv[0];
        const double c1 = Sv[1], c2 = Sv[1] + Sv[2];
        const double ssum = exp(BETA * (c1 - LAMDA)) + exp(BETA * (c2 - LAMDA));
        const double term = (1.0 / ALPHA) * log1p(exp(-ALPHA * (smm1 - LAMDA)))
                          + (1.0 / BETA)  * log1p(ssum);
        out[0] = (float)(3.0 * term);   // (b-1) * term, b = 4
    }
}

extern "C" void kernel_launch(void* const* d_in, const int* in_sizes, int n_in,
                              void* d_out, int out_size, void* d_ws, size_t ws_size,
                              hipStream_t stream)
{
    const float* fi = (const float*)d_in[0];
    const float* fj = (const float*)d_in[1];
    // d_in[2] is b == 4 (matches the compiled-in view count).
    float* partials = (float*)d_ws;                 // NVIEW * 1024 floats = 12 KB scratch
    const int nwg = B_ / ROWS_PER_WG;               // 128
    dim3 grid(nwg, NVIEW);
    attn_svm_kernel<<<grid, WAVES * 32, 0, stream>>>(fi, fj, partials, nwg);
    reduce_finalize_kernel<<<1, 1024, 0, stream>>>(partials, (float*)d_out, nwg * WAVES * 2);
}